// MLP_RL_11948599017964
// MI455X (gfx1250) — compile-verified
//
#include <hip/hip_runtime.h>

// MI455X / gfx1250 — wave32, WMMA. Fused hypernetwork MLP.
//
// Stage A (per node n):  tmp = eb[:, :, n, :]@W1+b1 ; out_spa = leaky(tmp @ Wspa[n] + bspa[n])
//   Wspa[n] generated in LDS from node_eb[n] (rank-16 contraction; pool stays in L2).
// Stage B (per b,t):     out_tem = leaky(out_spa[b,t] @ Wtem[b,t] + btem) ; logits = out_tem@W3+b3
//   Wtem generated in LDS from time_eb[b,t]; final 128->2 projection fused in epilogue.
//
// All matmuls use V_WMMA_F32_16X16X4_F32 (fp32-exact, matches fp32 reference).
// B-operand matrices live in LDS in K-pair-interleaved layout so a lane's B
// fragment {B[k][c], B[k+1][c]} is one aligned ds_load_b64 straight into the
// even-aligned VGPR pair the WMMA consumes (no v_mov packing).

typedef __attribute__((ext_vector_type(2))) float v2f;
typedef __attribute__((ext_vector_type(8))) float v8f;

#define B_    32
#define T_    24
#define N_    512
#define DIN   64
#define DOUT  2
#define HID   128
#define EMB   16
#define BT    (B_ * T_)          // 768
#define LDST  130                // staging row stride (floats): 8B-aligned, bank-clean A reads
#define PSTR  (2 * HID + 4)     // 260: K-pair row stride (floats), 8B aligned

__device__ __forceinline__ float leaky(float x) { return x >= 0.f ? x : 0.01f * x; }

__device__ __forceinline__ v8f wmma_f32(v2f a, v2f b, v8f c) {
    // 8 args: (neg_a, A, neg_b, B, c_mod, C, reuse_a, reuse_b)
    return __builtin_amdgcn_wmma_f32_16x16x4_f32(false, a, false, b, (short)0, c,
                                                 false, false);
}

// ---------------------------------------------------------------------------
// Kernel A: one workgroup per node n (512 blocks x 256 threads = 8 waves)
// ---------------------------------------------------------------------------
__global__ __launch_bounds__(256) void spatial_kernel(
    const float* __restrict__ eb,            // [BT, N, DIN]
    const float* __restrict__ node_eb,       // [N, EMB]
    const float* __restrict__ W1,            // [DIN, HID]
    const float* __restrict__ b1,            // [HID]
    const float* __restrict__ pool_spa,      // [EMB, HID, HID]
    const float* __restrict__ bias_pool_spa, // [EMB, HID]
    float* __restrict__ out_spa)             // [BT, N, HID]
{
    extern __shared__ float smem[];
    float* sW1  = smem;                         // (DIN/2) x PSTR  K-pair layout
    float* sWs  = sW1  + (DIN / 2) * PSTR;      // (HID/2) x PSTR  K-pair layout
    float* sTmp = sWs  + (HID / 2) * PSTR;      // 8 waves x 16 x LDST staging
    float* sB1  = sTmp + 8 * 16 * LDST;         // HID
    float* sBs  = sB1  + HID;                   // HID

    const int tid  = threadIdx.x;
    const int lane = tid & 31;
    const int wave = tid >> 5;
    const int n    = blockIdx.x;

    float ne[EMB];
#pragma unroll
    for (int d = 0; d < EMB; ++d) ne[d] = node_eb[n * EMB + d];

    // stage W1 into LDS, K-pair interleaved: sW1[(k/2)*PSTR + c*2 + (k&1)]
    for (int idx = tid; idx < DIN * HID; idx += 256) {
        int k = idx >> 7, c = idx & 127;
        sW1[(k >> 1) * PSTR + c * 2 + (k & 1)] = W1[idx];
    }
    if (tid < HID) sB1[tid] = b1[tid];

    // generate Wspa[n][i][o] = sum_d ne[d] * pool_spa[d][i][o]  (pool stays in L2)
    for (int idx = tid; idx < HID * HID; idx += 256) {
        int i = idx >> 7, o = idx & 127;
        float acc = 0.f;
#pragma unroll
        for (int d = 0; d < EMB; ++d)
            acc += ne[d] * pool_spa[(size_t)d * HID * HID + idx];
        sWs[(i >> 1) * PSTR + o * 2 + (i & 1)] = acc;
    }
    if (tid < HID) {
        float acc = 0.f;
#pragma unroll
        for (int d = 0; d < EMB; ++d) acc += ne[d] * bias_pool_spa[d * HID + tid];
        sBs[tid] = acc;
    }
    __syncthreads();

    const int mrow = lane & 15;   // A-fragment row / B,C column-in-tile
    const int kh   = lane >> 4;   // half-wave selects K pair (A/B) or M+8 (C)
    float* myTmp = sTmp + wave * 16 * LDST;

    for (int rt = wave; rt < BT / 16; rt += 8) {
        const int r0 = rt * 16;   // 16 (b,t)-rows per tile
        const float* arow = eb + ((size_t)(r0 + mrow) * N_ + n) * DIN;

        // ---- GEMM1: tmp[16x128] = ebtile[16x64] @ W1 -----------------------
        v8f acc[8] = {};
#pragma unroll 2
        for (int ks = 0; ks < DIN / 4; ++ks) {
            const int p = 2 * ks + kh;           // K-pair index: k0 = 2p
            v2f a = *(const v2f*)(arow + 2 * p);
            const float* brow = sW1 + p * PSTR + mrow * 2;
#pragma unroll
            for (int ct = 0; ct < 8; ++ct) {
                v2f b = *(const v2f*)(brow + ct * 32);   // one ds_load_b64
                acc[ct] = wmma_f32(a, b, acc[ct]);
            }
        }
        // C-layout -> LDS staging (+bias); re-read below in A-layout
#pragma unroll
        for (int ct = 0; ct < 8; ++ct) {
            const int col = ct * 16 + mrow;
#pragma unroll
            for (int j = 0; j < 8; ++j)
                myTmp[(j + 8 * kh) * LDST + col] = acc[ct][j] + sB1[col];
        }

        // ---- GEMM2: out[16x128] = leaky(tmp @ Wspa + bspa) -----------------
        v8f acc2[8] = {};
#pragma unroll 2
        for (int ks = 0; ks < HID / 4; ++ks) {
            const int p = 2 * ks + kh;
            v2f a = *(const v2f*)(myTmp + mrow * LDST + 2 * p);
            const float* brow = sWs + p * PSTR + mrow * 2;
#pragma unroll
            for (int ct = 0; ct < 8; ++ct) {
                v2f b = *(const v2f*)(brow + ct * 32);
                acc2[ct] = wmma_f32(a, b, acc2[ct]);
            }
        }
#pragma unroll
        for (int ct = 0; ct < 8; ++ct) {
            const int col = ct * 16 + mrow;
#pragma unroll
            for (int j = 0; j < 8; ++j) {
                const int r = r0 + j + 8 * kh;
                out_spa[((size_t)r * N_ + n) * HID + col] = leaky(acc2[ct][j] + sBs[col]);
            }
        }
    }
}

// ---------------------------------------------------------------------------
// Kernel B: one workgroup per (b,t) (768 blocks x 256 threads = 8 waves)
// ---------------------------------------------------------------------------
__global__ __launch_bounds__(256) void temporal_kernel(
    const float* __restrict__ out_spa,       // [BT, N, HID]
    const float* __restrict__ time_eb,       // [BT, EMB]
    const float* __restrict__ pool_tem,      // [EMB, HID, HID]
    const float* __restrict__ bias_pool_tem, // [EMB, HID]
    const float* __restrict__ W3,            // [HID, DOUT]
    const float* __restrict__ b3,            // [DOUT]
    float* __restrict__ logits)              // [BT, N, DOUT]
{
    extern __shared__ float smem[];
    float* sWt  = smem;                         // (HID/2) x PSTR  K-pair layout
    float* sOut = sWt  + (HID / 2) * PSTR;      // 8 waves x 16 x LDST staging
    float* sW3  = sOut + 8 * 16 * LDST;         // HID x DOUT
    float* sBt  = sW3  + HID * DOUT;            // HID
    float* sB3  = sBt  + HID;                   // DOUT

    const int tid  = threadIdx.x;
    const int lane = tid & 31;
    const int wave = tid >> 5;
    const int bt   = blockIdx.x;

    float te[EMB];
#pragma unroll
    for (int d = 0; d < EMB; ++d) te[d] = time_eb[bt * EMB + d];

    for (int idx = tid; idx < HID * HID; idx += 256) {
        int i = idx >> 7, o = idx & 127;
        float acc = 0.f;
#pragma unroll
        for (int d = 0; d < EMB; ++d)
            acc += te[d] * pool_tem[(size_t)d * HID * HID + idx];
        sWt[(i >> 1) * PSTR + o * 2 + (i & 1)] = acc;
    }
    if (tid < HID) {
        float acc = 0.f;
#pragma unroll
        for (int d = 0; d < EMB; ++d) acc += te[d] * bias_pool_tem[d * HID + tid];
        sBt[tid] = acc;
    }
    if (tid < HID * DOUT) sW3[tid] = W3[tid];
    if (tid < DOUT) sB3[tid] = b3[tid];
    __syncthreads();

    const int mrow = lane & 15;
    const int kh   = lane >> 4;
    float* myOut = sOut + wave * 16 * LDST;

    for (int rt = wave; rt < N_ / 16; rt += 8) {
        const int n0 = rt * 16;
        const float* arow = out_spa + ((size_t)bt * N_ + n0 + mrow) * HID;

        v8f acc[8] = {};
#pragma unroll 2
        for (int ks = 0; ks < HID / 4; ++ks) {
            const int p = 2 * ks + kh;
            v2f a = *(const v2f*)(arow + 2 * p);
            const float* brow = sWt + p * PSTR + mrow * 2;
#pragma unroll
            for (int ct = 0; ct < 8; ++ct) {
                v2f b = *(const v2f*)(brow + ct * 32);
                acc[ct] = wmma_f32(a, b, acc[ct]);
            }
        }
        // out_tem tile (leaky + bias) -> LDS, then fused 128->2 projection
#pragma unroll
        for (int ct = 0; ct < 8; ++ct) {
            const int col = ct * 16 + mrow;
#pragma unroll
            for (int j = 0; j < 8; ++j)
                myOut[(j + 8 * kh) * LDST + col] = leaky(acc[ct][j] + sBt[col]);
        }
        // 32 lanes -> 16 rows x 2 out-cols (W3 reads broadcast per half-pair)
        const int lr = lane >> 1;
        const int oc = lane & 1;
        float dot = sB3[oc];
#pragma unroll 8
        for (int k = 0; k < HID; ++k)
            dot += myOut[lr * LDST + k] * sW3[k * DOUT + oc];
        logits[((size_t)bt * N_ + n0 + lr) * DOUT + oc] = dot;
    }
}

// ---------------------------------------------------------------------------
extern "C" void kernel_launch(void* const* d_in, const int* in_sizes, int n_in,
                              void* d_out, int out_size, void* d_ws, size_t ws_size,
                              hipStream_t stream) {
    (void)in_sizes; (void)n_in; (void)out_size; (void)ws_size;
    const float* eb            = (const float*)d_in[0];
    const float* time_eb       = (const float*)d_in[1];
    const float* node_eb       = (const float*)d_in[2];
    const float* W1            = (const float*)d_in[3];
    const float* b1            = (const float*)d_in[4];
    const float* W3            = (const float*)d_in[5];
    const float* b3            = (const float*)d_in[6];
    const float* pool_spa      = (const float*)d_in[7];
    const float* bias_pool_spa = (const float*)d_in[8];
    const float* pool_tem      = (const float*)d_in[9];
    const float* bias_pool_tem = (const float*)d_in[10];
    float* logits  = (float*)d_out;
    float* out_spa = (float*)d_ws;   // BT*N*HID fp32 = 201 MB scratch

    const size_t lds_a = (size_t)((DIN / 2) * PSTR + (HID / 2) * PSTR +
                                  8 * 16 * LDST + 2 * HID) * sizeof(float);   // ~167 KB
    const size_t lds_b = (size_t)((HID / 2) * PSTR + 8 * 16 * LDST +
                                  HID * DOUT + HID + DOUT) * sizeof(float);   // ~135 KB

    spatial_kernel<<<N_, 256, lds_a, stream>>>(eb, node_eb, W1, b1, pool_spa,
                                               bias_pool_spa, out_spa);
    temporal_kernel<<<BT, 256, lds_b, stream>>>(out_spa, time_eb, pool_tem,
                                                bias_pool_tem, W3, b3, logits);
}